// HebbianLayer_64914135712120
// MI455X (gfx1250) — compile-verified
//
#include <hip/hip_runtime.h>
#include <hip/hip_bf16.h>
#include <math.h>

#define D 4096
#define ALPHA 0.001f
#define GAMMA 0.99f
#define EPSF  0.0001f

typedef float v2f __attribute__((ext_vector_type(2)));
typedef float v8f __attribute__((ext_vector_type(8)));

// b128 async payload type per clang diagnostic: int __attribute__((vector_size(16)))
typedef int v4i_vs __attribute__((__vector_size__(16)));
typedef __attribute__((address_space(1))) v4i_vs as1_v4i;
typedef __attribute__((address_space(3))) v4i_vs as3_v4i;

#if defined(__has_builtin)
#  if __has_builtin(__builtin_amdgcn_global_load_async_to_lds_b128)
#    define HAVE_ASYNC_LDS 1
#  endif
#endif

#if defined(HAVE_ASYNC_LDS)
#  if __has_builtin(__builtin_amdgcn_s_wait_asynccnt)
#    define WAIT_ASYNC(N) __builtin_amdgcn_s_wait_asynccnt(N)
#  else
#    define WAIT_ASYNC(N) asm volatile("s_wait_asynccnt %0" ::"i"(N) : "memory")
#  endif
#  define NBUF 2
#else
#  define NBUF 1
#endif

#define TSTRIDE 36   // floats; rows 144B -> 16B aligned, bank-conflict-free B reads

// ---------------------------------------------------------------------------
// K1: u_part[kseg][row] = W[row, kseg*1024:(kseg+1)*1024] . x[seg]
// One wave = 16-row tile over a 1024-wide K segment, V_WMMA_F32_16X16X4_F32,
// A = broadcast(x), B = W^T tile staged in LDS (async double-buffered).
// ---------------------------------------------------------------------------
__global__ __launch_bounds__(128) void hebb_gemv_wmma(
    const float* __restrict__ W, const float* __restrict__ x,
    float* __restrict__ u_part) {
  __shared__ __align__(16) float xsh[D];
  __shared__ __align__(16) float tile[4][NBUF][16 * TSTRIDE];

  const int tid  = threadIdx.x;
  const int lane = tid & 31;
  const int wave = tid >> 5;

  // vectorized x preload: 1024 float4s over 128 threads
#pragma unroll
  for (int it = 0; it < 8; ++it) {
    const int f = it * 128 + tid;
    reinterpret_cast<float4*>(xsh)[f] = reinterpret_cast<const float4*>(x)[f];
  }

  const int gw       = blockIdx.x * 4 + wave;   // 0..1023
  const int row_tile = gw >> 2;                 // 0..255
  const int kseg     = gw & 3;                  // 0..3
  const int rowbase  = row_tile * 16;
  const int kbeg     = kseg * (D / 4);
  const int kend     = kbeg + (D / 4);

  const int half = lane >> 4;
  const int n16  = lane & 15;

  v8f acc = {};

  // stage one 16x32 f32 tile of W into LDS (4 x b128 per lane)
  auto issue_tile = [&](float* dstbase, int k0) {
#pragma unroll
    for (int it = 0; it < 4; ++it) {
      const int idx = it * 32 + lane;          // 0..127
      const int r   = idx >> 3;                // 0..15
      const int c4  = (idx & 7) << 2;          // 0,4,...,28
      const float* src = W + (size_t)(rowbase + r) * D + k0 + c4;
      float*       dst = dstbase + r * TSTRIDE + c4;
#if defined(HAVE_ASYNC_LDS)
      __builtin_amdgcn_global_load_async_to_lds_b128(
          (as1_v4i*)src, (as3_v4i*)dst, 0, 0);
#else
      const float4 v = *reinterpret_cast<const float4*>(src);
      dst[0] = v.x; dst[1] = v.y; dst[2] = v.z; dst[3] = v.w;
#endif
    }
  };

  auto compute_tile = [&](const float* mt, int k0) {
#pragma unroll
    for (int kk = 0; kk < 8; ++kk) {
      const int kq = (kk << 2) + (half << 1);  // K for this lane's VGPR0
      v2f a, b;
      a[0] = xsh[k0 + kq];                     // A[m,k] = x[k] (all m)
      a[1] = xsh[k0 + kq + 1];
      b[0] = mt[n16 * TSTRIDE + kq];           // B[k,n] = W[rowbase+n, k0+k]
      b[1] = mt[n16 * TSTRIDE + kq + 1];
      acc = __builtin_amdgcn_wmma_f32_16x16x4_f32(
          false, a, false, b, (short)0, acc, false, false);
    }
  };

  __syncthreads();   // xsh visible to all waves

#if defined(HAVE_ASYNC_LDS)
  // double-buffered async pipeline; each wave owns its tile buffers
  issue_tile(&tile[wave][0][0], kbeg);
  int cur = 0;
  for (int k0 = kbeg; k0 < kend; k0 += 32) {
    if (k0 + 32 < kend) {
      issue_tile(&tile[wave][cur ^ 1][0], k0 + 32);
      WAIT_ASYNC(4);                 // oldest 4 (current tile) complete
    } else {
      WAIT_ASYNC(0);
    }
    asm volatile("" ::: "memory");
    compute_tile(&tile[wave][cur][0], k0);
    cur ^= 1;
  }
#else
  for (int k0 = kbeg; k0 < kend; k0 += 32) {
    issue_tile(&tile[wave][0][0], k0);
    __syncthreads();
    compute_tile(&tile[wave][0][0], k0);
    __syncthreads();
  }
#endif

  if (lane < 16) u_part[kseg * D + rowbase + lane] = acc[0];  // D[0, n]
}

// ---------------------------------------------------------------------------
// K2: single block: u = sum(partials)+bias; m = max(relu(u)); y = r^2/m^2;
// exp_avg update; mean; gf_pos/gf_neg.
// ---------------------------------------------------------------------------
__global__ __launch_bounds__(1024) void hebb_act(
    const float* __restrict__ u_part, const float* __restrict__ bias,
    const float* __restrict__ exp_avg,
    float* __restrict__ y_out, float* __restrict__ ea_out,
    float* __restrict__ y_ws, float* __restrict__ gfp, float* __restrict__ gfn) {
  __shared__ float red[1024];
  const int t = threadIdx.x;

  const float4 b4 = reinterpret_cast<const float4*>(bias)[t];
  const float4 p0 = reinterpret_cast<const float4*>(u_part)[t];
  const float4 p1 = reinterpret_cast<const float4*>(u_part + D)[t];
  const float4 p2 = reinterpret_cast<const float4*>(u_part + 2 * D)[t];
  const float4 p3 = reinterpret_cast<const float4*>(u_part + 3 * D)[t];
  const float4 e4 = reinterpret_cast<const float4*>(exp_avg)[t];

  float r[4], eav[4];
  r[0] = fmaxf(b4.x + p0.x + p1.x + p2.x + p3.x, 0.0f);
  r[1] = fmaxf(b4.y + p0.y + p1.y + p2.y + p3.y, 0.0f);
  r[2] = fmaxf(b4.z + p0.z + p1.z + p2.z + p3.z, 0.0f);
  r[3] = fmaxf(b4.w + p0.w + p1.w + p2.w + p3.w, 0.0f);
  eav[0] = e4.x; eav[1] = e4.y; eav[2] = e4.z; eav[3] = e4.w;

  red[t] = fmaxf(fmaxf(r[0], r[1]), fmaxf(r[2], r[3]));
  __syncthreads();
  for (int s = 512; s > 0; s >>= 1) {
    if (t < s) red[t] = fmaxf(red[t], red[t + s]);
    __syncthreads();
  }
  const float m = red[0];
  __syncthreads();

  const float inv_m2 = 1.0f / (m * m);
  float ys[4], eas[4], lsum = 0.0f;
#pragma unroll
  for (int i = 0; i < 4; ++i) {
    const float yy = r[i] * r[i] * inv_m2;
    ys[i]  = yy;
    eas[i] = GAMMA * eav[i] + (1.0f - GAMMA) * yy;
    lsum  += eas[i];
  }
  red[t] = lsum;
  __syncthreads();
  for (int s = 512; s > 0; s >>= 1) {
    if (t < s) red[t] = red[t] + red[t + s];
    __syncthreads();
  }
  const float inv_avg = (float)D / red[0];

  float gpv[4];
#pragma unroll
  for (int i = 0; i < 4; ++i) {
    const float a = eas[i] * inv_avg;
    gpv[i] = EPSF * tanhf(-EPSF * (a - 1.0f)) + 1.0f;
  }
  const float4 yo  = make_float4(ys[0], ys[1], ys[2], ys[3]);
  const float4 eo  = make_float4(eas[0], eas[1], eas[2], eas[3]);
  const float4 gp4 = make_float4(gpv[0], gpv[1], gpv[2], gpv[3]);
  const float4 gn4 = make_float4(1.0f / gpv[0], 1.0f / gpv[1],
                                 1.0f / gpv[2], 1.0f / gpv[3]);

  reinterpret_cast<float4*>(y_out)[t]  = yo;
  reinterpret_cast<float4*>(y_ws)[t]   = yo;
  reinterpret_cast<float4*>(ea_out)[t] = eo;
  reinterpret_cast<float4*>(gfp)[t]    = gp4;
  reinterpret_cast<float4*>(gfn)[t]    = gn4;
}

// ---------------------------------------------------------------------------
// K3a: chunk partials  S[c][j] = sum_{k in chunk c (128 rows)} y[k]*W[k,j]
// Each thread owns 4 columns (float4 per row): 512B per wave per load.
// ---------------------------------------------------------------------------
__global__ __launch_bounds__(256) void hebb_colsum(
    const float* __restrict__ W, const float* __restrict__ y,
    float* __restrict__ S) {
  const int chunk = blockIdx.x >> 2;                         // 0..31
  const int j4    = ((blockIdx.x & 3) << 10) + (threadIdx.x << 2);
  const int kb    = chunk << 7;
  float4 s = make_float4(0.f, 0.f, 0.f, 0.f);
  for (int k = 0; k < 128; ++k) {
    const float  yk = y[kb + k];
    const float4 w  = *reinterpret_cast<const float4*>(W + (size_t)(kb + k) * D + j4);
    s.x = fmaf(yk, w.x, s.x);
    s.y = fmaf(yk, w.y, s.y);
    s.z = fmaf(yk, w.z, s.z);
    s.w = fmaf(yk, w.w, s.w);
  }
  *reinterpret_cast<float4*>(S + chunk * D + j4) = s;
}

// ---------------------------------------------------------------------------
// K3b: per (chunk, 4 columns): carry-in + serial inclusive scan over 128 rows,
// Hebbian update + sign-dependent decay, write w2 (float4 stores).
// ---------------------------------------------------------------------------
__global__ __launch_bounds__(256) void hebb_update(
    const float* __restrict__ W, const float* __restrict__ x,
    const float* __restrict__ y, const float* __restrict__ gfp,
    const float* __restrict__ gfn, const float* __restrict__ S,
    float* __restrict__ w_out) {
  const int chunk = blockIdx.x >> 2;
  const int j4    = ((blockIdx.x & 3) << 10) + (threadIdx.x << 2);

  float4 s = make_float4(0.f, 0.f, 0.f, 0.f);
  for (int c = 0; c < chunk; ++c) {
    const float4 sv = *reinterpret_cast<const float4*>(S + c * D + j4);
    s.x += sv.x; s.y += sv.y; s.z += sv.z; s.w += sv.w;
  }

  const float4 xv = *reinterpret_cast<const float4*>(x + j4);
  const int kb = chunk << 7;
  for (int k = kb; k < kb + 128; ++k) {
    const float4 w  = *reinterpret_cast<const float4*>(W + (size_t)k * D + j4);
    const float  yk = y[k];
    const float  gp = gfp[k];
    const float  gn = gfn[k];
    s.x = fmaf(yk, w.x, s.x);                       // inclusive scan
    s.y = fmaf(yk, w.y, s.y);
    s.z = fmaf(yk, w.z, s.z);
    s.w = fmaf(yk, w.w, s.w);
    const float w1x = w.x + ALPHA * (yk * xv.x - yk * s.x);
    const float w1y = w.y + ALPHA * (yk * xv.y - yk * s.y);
    const float w1z = w.z + ALPHA * (yk * xv.z - yk * s.z);
    const float w1w = w.w + ALPHA * (yk * xv.w - yk * s.w);
    float4 o;
    o.x = (w1x > 0.f) ? w1x * gp : ((w1x < 0.f) ? w1x * gn : 0.f);
    o.y = (w1y > 0.f) ? w1y * gp : ((w1y < 0.f) ? w1y * gn : 0.f);
    o.z = (w1z > 0.f) ? w1z * gp : ((w1z < 0.f) ? w1z * gn : 0.f);
    o.w = (w1w > 0.f) ? w1w * gp : ((w1w < 0.f) ? w1w * gn : 0.f);
    *reinterpret_cast<float4*>(w_out + (size_t)k * D + j4) = o;
  }
}

// ---------------------------------------------------------------------------
extern "C" void kernel_launch(void* const* d_in, const int* in_sizes, int n_in,
                              void* d_out, int out_size, void* d_ws, size_t ws_size,
                              hipStream_t stream) {
  (void)in_sizes; (void)n_in; (void)out_size; (void)ws_size;
  const float* x    = (const float*)d_in[0];
  const float* W    = (const float*)d_in[1];
  const float* bias = (const float*)d_in[2];
  const float* ea   = (const float*)d_in[3];

  float* out    = (float*)d_out;
  float* y_out  = out;                        // [0, 4096)
  float* w_out  = out + D;                    // [4096, 4096 + D*D)
  float* ea_out = out + D + (size_t)D * D;    // last 4096

  float* ws     = (float*)d_ws;
  float* u_part = ws;            // 4*D floats
  float* y_ws   = ws + 4 * D;    // D
  float* gfp    = ws + 5 * D;    // D
  float* gfn    = ws + 6 * D;    // D
  float* S      = ws + 7 * D;    // 32*D

  hebb_gemv_wmma<<<256, 128, 0, stream>>>(W, x, u_part);
  hebb_act<<<1, 1024, 0, stream>>>(u_part, bias, ea, y_out, ea_out, y_ws, gfp, gfn);
  hebb_colsum<<<128, 256, 0, stream>>>(W, y_ws, S);
  hebb_update<<<128, 256, 0, stream>>>(W, x, y_ws, gfp, gfn, S, w_out);
}